// SharedGroupLinearLayer_8744553415250
// MI455X (gfx1250) — compile-verified
//
#include <hip/hip_runtime.h>

typedef __attribute__((ext_vector_type(2))) float v2f;
typedef __attribute__((ext_vector_type(4))) float v4f;
typedef __attribute__((ext_vector_type(8))) float v8f;

#define NBLK 16   // blocks per token group (= tokens per tile)
#define DIN  64
#define DOUT 64
#define NT   2
#define ROWP 68   // padded LDS row stride (64 + 4) -> conflict-free, 16B aligned
#define WPB  4    // waves per block
#define TPB  (WPB * 32)

__global__ __launch_bounds__(TPB) void sgll_wmma_f32(
    const float* __restrict__ x,
    const float* __restrict__ emb,
    const float* __restrict__ rw,    // (DIN, NT) flat: d*NT + t
    const float* __restrict__ wst,   // (NT, DOUT, DIN)
    const float* __restrict__ bst,   // (NT, DOUT)
    float* __restrict__ out,
    int nTiles)
{
    __shared__ float ws[NT][DOUT * ROWP];   // padded weight rows (W[t][o][d])
    __shared__ float xs[WPB][16 * ROWP];    // per-wave padded x tile
    __shared__ float embS[NBLK * DIN];
    __shared__ float rwT[NT * DIN];         // transposed read_w: t*DIN + d
    __shared__ float bS[NT * DOUT];
    __shared__ float attS[WPB][32];

    const int tid = threadIdx.x;

    // ---- one-time staging (whole workgroup) ----
    for (int i = tid; i < NT * DOUT * DIN; i += TPB) {
        int t = i / (DOUT * DIN);
        int r = (i / DIN) % DOUT;
        int d = i % DIN;
        ws[t][r * ROWP + d] = wst[i];
    }
    for (int i = tid; i < NBLK * DIN; i += TPB) embS[i] = emb[i];
    for (int i = tid; i < DIN * NT; i += TPB)
        rwT[(i % NT) * DIN + (i / NT)] = rw[i];
    for (int i = tid; i < NT * DOUT; i += TPB) bS[i] = bst[i];
    __syncthreads();

    const int wv   = tid >> 5;
    const int lane = tid & 31;
    const int half = lane >> 4;   // 0: lanes 0-15, 1: lanes 16-31
    const int mrow = lane & 15;
    float* xw = xs[wv];           // wave-private LDS tile

    const int wave0 = blockIdx.x * WPB + wv;
    const int wstep = gridDim.x * WPB;

    for (int tile = wave0; tile < nTiles; tile += wstep) {
        const float* xg = x + (size_t)tile * (NBLK * DIN);

        // ---- stage x + emb into padded LDS (coalesced b128 loads) ----
        #pragma unroll
        for (int i = 0; i < 8; ++i) {
            int idx = (i * 32 + lane) * 4;          // 0..1023, step 4
            v4f v = *(const v4f*)(xg + idx);
            v4f e = *(const v4f*)(embS + idx);
            v += e;
            int r = idx >> 6;                       // row (=block) 0..15
            int c = idx & 63;
            *(v4f*)(xw + r * ROWP + c) = v;
        }
        asm volatile("s_wait_dscnt 0" ::: "memory");

        // ---- A fragments: 16x4 f32 tiles. lane m = row, VGPR0/1 = K pair,
        //      halves hold K+0..1 / K+2..3 (ISA 7.12.2 layout) ----
        v2f afrag[16];
        #pragma unroll
        for (int kk = 0; kk < 16; ++kk)
            afrag[kk] = *(const v2f*)(xw + mrow * ROWP + kk * 4 + 2 * half);

        // ---- routing logits: lane = (token mrow, template half) ----
        float lg = 0.f;
        #pragma unroll
        for (int q = 0; q < 16; ++q) {
            v4f xv = *(const v4f*)(xw + mrow * ROWP + q * 4);
            v4f rv = *(const v4f*)(rwT + half * DIN + q * 4);
            lg += xv.x * rv.x + xv.y * rv.y + xv.z * rv.z + xv.w * rv.w;
        }
        float lo  = __shfl_xor(lg, 16, 32);                 // other template's logit
        float att = 1.0f / (1.0f + __expf(lo - lg));        // softmax over 2
        attS[wv][mrow * NT + half] = att;
        asm volatile("s_wait_dscnt 0" ::: "memory");

        // att arranged for C/D layout: VGPR j holds token M = j + 8*half
        float a0[8], a1[8];
        #pragma unroll
        for (int j = 0; j < 8; ++j) {
            a0[j] = attS[wv][(j + 8 * half) * NT + 0];
            a1[j] = attS[wv][(j + 8 * half) * NT + 1];
        }

        float* og = out + (size_t)tile * (NBLK * DOUT);

        #pragma unroll
        for (int nt = 0; nt < 4; ++nt) {
            // bias folded into C operand (same value for all 8 M-rows of a lane)
            float b0 = bS[0 * DOUT + nt * 16 + mrow];
            float b1 = bS[1 * DOUT + nt * 16 + mrow];
            v8f acc0 = {b0, b0, b0, b0, b0, b0, b0, b0};
            v8f acc1 = {b1, b1, b1, b1, b1, b1, b1, b1};

            // B fragment: lane n = column (output o = nt*16 + mrow), K split by half.
            // B[k][n] = W[o][k]  (out = xf @ W^T)
            const float* w0 = &ws[0][(nt * 16 + mrow) * ROWP + 2 * half];
            const float* w1 = &ws[1][(nt * 16 + mrow) * ROWP + 2 * half];

            #pragma unroll
            for (int kk = 0; kk < 16; ++kk) {
                v2f bf0 = *(const v2f*)(w0 + kk * 4);
                v2f bf1 = *(const v2f*)(w1 + kk * 4);
                // two independent accumulation chains keep the XDL pipelined
                acc0 = __builtin_amdgcn_wmma_f32_16x16x4_f32(
                    false, afrag[kk], false, bf0, (short)0, acc0, false, false);
                acc1 = __builtin_amdgcn_wmma_f32_16x16x4_f32(
                    false, afrag[kk], false, bf1, (short)0, acc1, false, false);
            }

            // combine templates with att and store:
            // VGPR j, lane -> token m = j + 8*half, column n = nt*16 + mrow
            #pragma unroll
            for (int j = 0; j < 8; ++j) {
                float o = a0[j] * acc0[j] + a1[j] * acc1[j];
                og[(j + 8 * half) * DOUT + nt * 16 + mrow] = o;
            }
        }
    }
}

extern "C" void kernel_launch(void* const* d_in, const int* in_sizes, int n_in,
                              void* d_out, int out_size, void* d_ws, size_t ws_size,
                              hipStream_t stream) {
    const float* x   = (const float*)d_in[0];  // (T, B, NB*DIN)
    const float* emb = (const float*)d_in[1];  // (1, 1, NB*DIN)
    const float* rw  = (const float*)d_in[2];  // (1, DIN, NT)
    const float* wst = (const float*)d_in[3];  // (NT, DOUT, DIN)
    const float* bst = (const float*)d_in[4];  // (NT, DOUT)
    float* out = (float*)d_out;

    int nTiles = in_sizes[0] / (NBLK * DIN);   // one tile = one (t,b) row = 16 tokens

    int blocks = 2048;                          // 8192 waves -> 2 tiles/wave
    int maxBlocks = (nTiles + WPB - 1) / WPB;
    if (blocks > maxBlocks) blocks = maxBlocks;

    sgll_wmma_f32<<<dim3(blocks), dim3(TPB), 0, stream>>>(
        x, emb, rw, wst, bst, out, nTiles);
}